// LFQQuantizerEMA_KMeans_25409026523971
// MI455X (gfx1250) — compile-verified
//
#include <hip/hip_runtime.h>
#include <hip/hip_bf16.h>

// ---------------------------------------------------------------------------
// LFQ / VQ-VAE EMA training step on MI455X (gfx1250, wave32, WMMA).
// Heavy parts (z->codebook distances, codebook->codebook distances) use
// V_WMMA_F32_16X16X4_F32: full f32 precision (argmin-safe) on the matrix pipe.
// ---------------------------------------------------------------------------

#define NUM_CODES 8192
#define CODE_DIM 64
#define ROWS (4 * 4096)          // B*S = 16384
#define DECAY 0.99f
#define EPS 1e-5f
#define DEAD_THRESHOLD 5.0f

typedef __attribute__((ext_vector_type(2))) float v2f;
typedef __attribute__((ext_vector_type(8))) float v8f;

// --------------------------------------------------------------------------
// zero scratch accumulators
// --------------------------------------------------------------------------
__global__ void zero_kernel(float* p, int n) {
    int t = blockIdx.x * blockDim.x + threadIdx.x;
    if (t < n) p[t] = 0.0f;
}

// --------------------------------------------------------------------------
// per-row sum of squares (rows of length 64)
// --------------------------------------------------------------------------
__global__ void sumsq_kernel(const float* __restrict__ X, float* __restrict__ out, int rows) {
    int r = blockIdx.x * blockDim.x + threadIdx.x;
    if (r >= rows) return;
    const float* p = X + (size_t)r * CODE_DIM;
    float acc = 0.0f;
#pragma unroll
    for (int k = 0; k < CODE_DIM; ++k) acc += p[k] * p[k];
    out[r] = acc;
}

// --------------------------------------------------------------------------
// WMMA f32 distance + argmin kernel (templated on column-mask presence).
//   Q: M x 64 rows, C: N x 64 codes.
//   d2[m][n] = qsq[m] + csq[n] (+maskAdd[n]) - 2 * Q[m].C[n]
//   outIdx[m] = argmin_n d2[m][n]   (ties -> smallest n, like jnp.argmin)
// One wave (32 lanes) owns a 16-row tile. A tile lives in 32 VGPRs/lane and
// is reused across all N/16 B tiles. Two n-tiles per iteration with four
// accumulators: dependent WMMAs are 4 apart and 4 b64 loads fly per clause.
// --------------------------------------------------------------------------
template <bool HAS_MASK>
__launch_bounds__(128)
__global__ void argmin_dist_kernel(const float* __restrict__ Q,
                                   const float* __restrict__ C,
                                   const float* __restrict__ qsq,
                                   const float* __restrict__ csq,
                                   const float* __restrict__ maskAdd,
                                   int N,
                                   int* __restrict__ outIdx) {
    const int lane = threadIdx.x & 31;
    const int wave = threadIdx.x >> 5;
    const int lo = lane & 15;   // n (or m for A rows) within tile
    const int hi = lane >> 4;   // half-wave select
    const int m0 = (blockIdx.x * 4 + wave) * 16;

    // A tile: lane holds Q[m0+lo][k = 4*kk + 2*hi + {0,1}], kk = 0..15
    const float* qrow = Q + (size_t)(m0 + lo) * CODE_DIM + 2 * hi;
    v2f a[16];
#pragma unroll
    for (int kk = 0; kk < 16; ++kk)
        a[kk] = *(const v2f*)(qrow + 4 * kk);

    // qsq values this lane finishes: rows m = 8*hi + j
    float qs[8];
#pragma unroll
    for (int j = 0; j < 8; ++j) qs[j] = qsq[m0 + 8 * hi + j];

    float best[8];
    int bidx[8];
#pragma unroll
    for (int j = 0; j < 8; ++j) { best[j] = 3.4e38f; bidx[j] = 0; }

    for (int n0 = 0; n0 < N; n0 += 32) {
        const float* crow0 = C + (size_t)(n0 + lo) * CODE_DIM + 2 * hi;
        const float* crow1 = crow0 + 16 * CODE_DIM;
        __builtin_prefetch(crow0 + 32 * CODE_DIM, 0, 3);  // tiles two steps ahead

        v8f acc0 = {};
        v8f acc1 = {};
        v8f acc2 = {};
        v8f acc3 = {};
#pragma unroll
        for (int kk = 0; kk < 16; kk += 2) {
            v2f b0 = *(const v2f*)(crow0 + 4 * kk);
            v2f b1 = *(const v2f*)(crow0 + 4 * (kk + 1));
            v2f b2 = *(const v2f*)(crow1 + 4 * kk);
            v2f b3 = *(const v2f*)(crow1 + 4 * (kk + 1));
            acc0 = __builtin_amdgcn_wmma_f32_16x16x4_f32(false, a[kk], false, b0,
                                                         (short)0, acc0, false, false);
            acc1 = __builtin_amdgcn_wmma_f32_16x16x4_f32(false, a[kk + 1], false, b1,
                                                         (short)0, acc1, false, false);
            acc2 = __builtin_amdgcn_wmma_f32_16x16x4_f32(false, a[kk], false, b2,
                                                         (short)0, acc2, false, false);
            acc3 = __builtin_amdgcn_wmma_f32_16x16x4_f32(false, a[kk + 1], false, b3,
                                                         (short)0, acc3, false, false);
        }

        float cn0 = csq[n0 + lo];
        float cn1 = csq[n0 + 16 + lo];
        if (HAS_MASK) {
            cn0 += maskAdd[n0 + lo];
            cn1 += maskAdd[n0 + 16 + lo];
        }
        const int idx0 = n0 + lo;
        const int idx1 = n0 + 16 + lo;
#pragma unroll
        for (int j = 0; j < 8; ++j) {
            float d0 = qs[j] + cn0 - 2.0f * (acc0[j] + acc1[j]);
            float d1 = qs[j] + cn1 - 2.0f * (acc2[j] + acc3[j]);
            if (d0 < best[j]) { best[j] = d0; bidx[j] = idx0; }  // strict <: keeps first n
            if (d1 < best[j]) { best[j] = d1; bidx[j] = idx1; }
        }
    }

    // reduce over the 16 lanes holding different n for the same rows
#pragma unroll
    for (int s = 8; s >= 1; s >>= 1) {
#pragma unroll
        for (int j = 0; j < 8; ++j) {
            float od = __shfl_xor(best[j], s, 32);
            int oi = __shfl_xor(bidx[j], s, 32);
            if (od < best[j] || (od == best[j] && oi < bidx[j])) { best[j] = od; bidx[j] = oi; }
        }
    }
    if (lo == 0) {
#pragma unroll
        for (int j = 0; j < 8; ++j) outIdx[m0 + 8 * hi + j] = bidx[j];
    }
}

// --------------------------------------------------------------------------
// z_q gather (from PRE-update codebook), index write, cluster/embed scatter
// --------------------------------------------------------------------------
__global__ void gather_scatter_kernel(const float* __restrict__ z_e,
                                      const float* __restrict__ codebook,
                                      const int* __restrict__ idx,
                                      float* __restrict__ zq_out,
                                      float* __restrict__ idx_out,
                                      float* __restrict__ cluster,
                                      float* __restrict__ embed) {
    int t = blockIdx.x * blockDim.x + threadIdx.x;
    if (t >= ROWS * CODE_DIM) return;
    int r = t >> 6, d = t & 63;
    int k = idx[r];
    zq_out[t] = codebook[(size_t)k * CODE_DIM + d];
    atomicAdd(&embed[(size_t)k * CODE_DIM + d], z_e[t]);
    if (d == 0) {
        atomicAdd(&cluster[k], 1.0f);
        idx_out[r] = (float)k;
    }
}

// --------------------------------------------------------------------------
// EMA update: new_cs / new_cb
// --------------------------------------------------------------------------
__global__ void ema_kernel(const float* __restrict__ ema_cs,
                           const float* __restrict__ ema_cb,
                           const float* __restrict__ cluster,
                           const float* __restrict__ embed,
                           float* __restrict__ out_cs,
                           float* __restrict__ out_cb) {
    int t = blockIdx.x * blockDim.x + threadIdx.x;
    if (t >= NUM_CODES * CODE_DIM) return;
    out_cb[t] = ema_cb[t] * DECAY + (1.0f - DECAY) * embed[t];
    if ((t & 63) == 0) {
        int i = t >> 6;
        out_cs[i] = ema_cs[i] * DECAY + (1.0f - DECAY) * cluster[i];
    }
}

// single-workgroup deterministic sum of new_ema_cs
__global__ void reduce_total_kernel(const float* __restrict__ cs, float* __restrict__ n_total) {
    __shared__ float s[256];
    float acc = 0.0f;
    for (int i = threadIdx.x; i < NUM_CODES; i += 256) acc += cs[i];
    s[threadIdx.x] = acc;
    __syncthreads();
    for (int off = 128; off >= 1; off >>= 1) {
        if ((int)threadIdx.x < off) s[threadIdx.x] += s[threadIdx.x + off];
        __syncthreads();
    }
    if (threadIdx.x == 0) *n_total = s[0];
}

__global__ void normalize_kernel(const float* __restrict__ new_cb_ema,
                                 const float* __restrict__ new_cs,
                                 const float* __restrict__ n_total,
                                 float* __restrict__ cb_new) {
    int t = blockIdx.x * blockDim.x + threadIdx.x;
    if (t >= NUM_CODES * CODE_DIM) return;
    int i = t >> 6;
    float norm = (new_cs[i] + EPS) / (*n_total + (float)NUM_CODES * EPS);
    cb_new[t] = new_cb_ema[t] / norm;
}

// dead detection: maskAdd[n] = big if dead (column mask), alive counter
__global__ void dead_kernel(const float* __restrict__ usage,
                            const float* __restrict__ cluster,
                            float* __restrict__ maskAdd,
                            int* __restrict__ aliveCount) {
    int i = blockIdx.x * blockDim.x + threadIdx.x;
    if (i >= NUM_CODES) return;
    float u = usage[i] + cluster[i];
    bool dead = u < DEAD_THRESHOLD;
    maskAdd[i] = dead ? 1e30f : 0.0f;
    if (!dead) atomicAdd(aliveCount, 1);
}

__global__ void replace_kernel(const float* __restrict__ cb_new,
                               const float* __restrict__ maskAdd,
                               const int* __restrict__ nearest,
                               const int* __restrict__ aliveCount,
                               float* __restrict__ out_cb) {
    int t = blockIdx.x * blockDim.x + threadIdx.x;
    if (t >= NUM_CODES * CODE_DIM) return;
    int i = t >> 6, d = t & 63;
    bool dead = maskAdd[i] > 0.0f;
    bool anyAlive = (*aliveCount) > 0;
    out_cb[t] = (dead && anyAlive) ? cb_new[(size_t)nearest[i] * CODE_DIM + d] : cb_new[t];
}

// --------------------------------------------------------------------------
extern "C" void kernel_launch(void* const* d_in, const int* in_sizes, int n_in,
                              void* d_out, int out_size, void* d_ws, size_t ws_size,
                              hipStream_t stream) {
    const float* z_e      = (const float*)d_in[0];   // [16384,64]
    const float* codebook = (const float*)d_in[1];   // [8192,64]
    const float* ema_cs   = (const float*)d_in[2];   // [8192]
    const float* ema_cb   = (const float*)d_in[3];   // [8192,64]
    const float* usage    = (const float*)d_in[4];   // [8192]

    float* out = (float*)d_out;
    float* out_zq  = out;                                  // 16384*64
    float* out_idx = out + ROWS * CODE_DIM;                // 16384
    float* out_rep = out_idx + ROWS;                       // 8192*64
    float* out_cs  = out_rep + NUM_CODES * CODE_DIM;       // 8192
    float* out_cb  = out_cs + NUM_CODES;                   // 8192*64

    // workspace layout (floats); accumulators first so one zero pass covers them
    float* ws      = (float*)d_ws;
    float* cluster = ws;                                   // 8192
    float* embed   = ws + NUM_CODES;                       // 524288
    float* n_total = ws + NUM_CODES + NUM_CODES * CODE_DIM;        // 1
    int*   alive   = (int*)(n_total + 1);                           // 1
    float* zsq     = (float*)(alive + 1);                           // 16384
    float* csq_old = zsq + ROWS;                                    // 8192
    float* cb_new  = csq_old + NUM_CODES;                           // 524288 (8B aligned)
    float* cbsq    = cb_new + NUM_CODES * CODE_DIM;                 // 8192
    float* maskAdd = cbsq + NUM_CODES;                              // 8192
    int*   idx1    = (int*)(maskAdd + NUM_CODES);                   // 16384
    int*   nearest = idx1 + ROWS;                                   // 8192

    const int zeroCount = NUM_CODES + NUM_CODES * CODE_DIM + 2;     // cluster+embed+n_total+alive

    zero_kernel<<<(zeroCount + 255) / 256, 256, 0, stream>>>(ws, zeroCount);

    sumsq_kernel<<<(ROWS + 255) / 256, 256, 0, stream>>>(z_e, zsq, ROWS);
    sumsq_kernel<<<(NUM_CODES + 255) / 256, 256, 0, stream>>>(codebook, csq_old, NUM_CODES);

    // assignment: 16384 rows -> 1024 row-tiles -> 256 blocks of 4 waves
    argmin_dist_kernel<false><<<ROWS / 64, 128, 0, stream>>>(z_e, codebook, zsq, csq_old,
                                                             nullptr, NUM_CODES, idx1);

    gather_scatter_kernel<<<(ROWS * CODE_DIM + 255) / 256, 256, 0, stream>>>(
        z_e, codebook, idx1, out_zq, out_idx, cluster, embed);

    ema_kernel<<<(NUM_CODES * CODE_DIM + 255) / 256, 256, 0, stream>>>(
        ema_cs, ema_cb, cluster, embed, out_cs, out_cb);

    reduce_total_kernel<<<1, 256, 0, stream>>>(out_cs, n_total);

    normalize_kernel<<<(NUM_CODES * CODE_DIM + 255) / 256, 256, 0, stream>>>(
        out_cb, out_cs, n_total, cb_new);

    sumsq_kernel<<<(NUM_CODES + 255) / 256, 256, 0, stream>>>(cb_new, cbsq, NUM_CODES);

    dead_kernel<<<(NUM_CODES + 255) / 256, 256, 0, stream>>>(usage, cluster, maskAdd, alive);

    // nearest-alive per code: 8192 rows -> 512 tiles -> 128 blocks
    argmin_dist_kernel<true><<<NUM_CODES / 64, 128, 0, stream>>>(cb_new, cb_new, cbsq, cbsq,
                                                                 maskAdd, NUM_CODES, nearest);

    replace_kernel<<<(NUM_CODES * CODE_DIM + 255) / 256, 256, 0, stream>>>(
        cb_new, maskAdd, nearest, alive, out_rep);
}